// GlobalAttentionPool_31963146616865
// MI455X (gfx1250) — compile-verified
//
#include <hip/hip_runtime.h>
#include <hip/hip_bf16.h>

#define HIDDEN 128
#define EPS 1e-16f

typedef __attribute__((ext_vector_type(2))) float v2f;
typedef __attribute__((ext_vector_type(8))) float v8f;

// ---- order-preserving float <-> uint encoding for atomicMax-based segment max ----
__device__ __forceinline__ unsigned encf(float f) {
    unsigned u = __float_as_uint(f);
    return (u & 0x80000000u) ? ~u : (u | 0x80000000u);
}
__device__ __forceinline__ float decf(unsigned e) {
    unsigned u = (e & 0x80000000u) ? (e & 0x7fffffffu) : ~e;
    return __uint_as_float(u);
}

// K0: zero scratch + output; init per-graph max to enc(-FLT_MAX)
__global__ void gap_init(float* __restrict__ sval, unsigned* __restrict__ gmax,
                         float* __restrict__ denom, float* __restrict__ gx,
                         int n, int g) {
    int i = blockIdx.x * blockDim.x + threadIdx.x;
    if (i < n) sval[i] = 0.0f;
    if (i < g) { gmax[i] = 0x00800000u /* enc(-FLT_MAX) */; denom[i] = 0.0f; }
    if (i < g * HIDDEN) gx[i] = 0.0f;
}

// K1: WMMA projection: p_rel[i] = x_i . W_rel ; p_root[i] = x_i . W_root
// One wave handles 16 rows via a chain of 32x V_WMMA_F32_16X16X4_F32 over K=128.
// B tile lives in a pre-zero-padded LDS table so the inner loop is branch-free:
//   sB[p][col][j] = (col==0 ? Wrel : col==1 ? Wroot : 0)[2*p + j]
__global__ __launch_bounds__(256) void gap_proj(const float* __restrict__ x,
                                                const float* __restrict__ Wrel,
                                                const float* __restrict__ Wroot,
                                                float* __restrict__ prel,
                                                float* __restrict__ proot, int n) {
    __shared__ float sB[(HIDDEN / 2) * 16 * 2];   // 8 KB
    const int t = threadIdx.x;
    #pragma unroll
    for (int idx = t; idx < (HIDDEN / 2) * 16 * 2; idx += 256) {
        int j    = idx & 1;
        int colv = (idx >> 1) & 15;
        int p    = idx >> 5;
        int k    = 2 * p + j;
        float v  = (colv == 0) ? Wrel[k] : ((colv == 1) ? Wroot[k] : 0.0f);
        sB[idx] = v;
    }
    __syncthreads();

    const int lane    = t & 31;
    const int wave    = t >> 5;
    const int rowBase = blockIdx.x * 128 + wave * 16;
    const int m       = lane & 15;            // A-matrix row held by this lane
    const int col     = lane & 15;            // B/D column held by this lane
    const int half    = (lane < 16) ? 0 : 1;  // K sub-offset select (ISA 32-bit layout)
    const int koff    = half * 2;

    int row = rowBase + m;
    if (row >= n) row = n - 1;                // clamp: EXEC must stay all-1s for WMMA
    const float* xr = x + (size_t)row * HIDDEN + koff;
    const float* bp = sB + col * 2 + half * 32;

    // Preload all A fragments (32 x global_load_b64) so VMEM issues ahead of the
    // WMMA chain instead of a full wait before every matrix op.
    v2f a[HIDDEN / 4];
    #pragma unroll
    for (int i = 0; i < HIDDEN / 4; ++i) {
        a[i] = *(const v2f*)(xr + i * 4);
    }

    v8f c = {};
    #pragma unroll
    for (int i = 0; i < HIDDEN / 4; ++i) {
        v2f b = *(const v2f*)(bp + i * 64);   // unconditional ds_load_b64, imm offset
        c = __builtin_amdgcn_wmma_f32_16x16x4_f32(false, a[i], false, b,
                                                  (short)0, c, false, false);
    }

    // D layout: VGPR r holds (m = r + (lane<16 ? 0 : 8), n = lane&15).
    // Column 0 => p_rel, column 1 => p_root.
    if (col < 2) {
        float* outp = (col == 0) ? prel : proot;
        const int mbase = half * 8;
        #pragma unroll
        for (int r = 0; r < 8; ++r) {
            int rr = rowBase + mbase + r;
            if (rr < n) outp[rr] = c[r];
        }
    }
}

// K2: edge scatter of scalar messages: s_agg[dst] += p_rel[src]
__global__ void gap_edges(const int* __restrict__ src, const int* __restrict__ dst,
                          const float* __restrict__ prel, float* __restrict__ sval,
                          int e) {
    int i = blockIdx.x * blockDim.x + threadIdx.x;
    if (i < e) {
        atomicAdd(&sval[dst[i]], prel[src[i]]);
    }
}

// K3: s_i = agg_i + b_rel + p_root_i ; segment max via encoded atomicMax
__global__ void gap_score(float* __restrict__ sval, const float* __restrict__ proot,
                          const float* __restrict__ brel, const int* __restrict__ batch,
                          unsigned* __restrict__ gmax, int n) {
    int i = blockIdx.x * blockDim.x + threadIdx.x;
    if (i < n) {
        float s = sval[i] + proot[i] + brel[0];
        sval[i] = s;
        atomicMax(&gmax[batch[i]], encf(s));
    }
}

// K4: e_i = exp(s_i - max_g) ; denom_g += e_i   (sval is reused agg->s->e)
__global__ void gap_exp(float* __restrict__ sval, const int* __restrict__ batch,
                        const unsigned* __restrict__ gmax, float* __restrict__ denom,
                        int n) {
    int i = blockIdx.x * blockDim.x + threadIdx.x;
    if (i < n) {
        int g = batch[i];
        float ev = expf(sval[i] - decf(gmax[g]));
        sval[i] = ev;
        atomicAdd(&denom[g], ev);
    }
}

// K5: gx[g] += score_i * x_i. batch is sorted, so each block of 32 consecutive
// nodes accumulates per-feature in registers and flushes on graph boundary.
__global__ __launch_bounds__(HIDDEN) void gap_pool(const float* __restrict__ x,
                                                   const int* __restrict__ batch,
                                                   const float* __restrict__ ebuf,
                                                   const float* __restrict__ denom,
                                                   float* __restrict__ gx, int n) {
    const int h  = threadIdx.x;          // feature column 0..127
    const int i0 = blockIdx.x * 32;
    const int i1 = (i0 + 32 < n) ? (i0 + 32) : n;
    float acc = 0.0f;
    int   cur = -1;
    for (int i = i0; i < i1; ++i) {
        int g = batch[i];
        if (g != cur) {
            if (cur >= 0) atomicAdd(&gx[cur * HIDDEN + h], acc);
            cur = g;
            acc = 0.0f;
        }
        float sc = ebuf[i] / (denom[g] + EPS);
        acc = fmaf(x[(size_t)i * HIDDEN + h], sc, acc);
    }
    if (cur >= 0) atomicAdd(&gx[cur * HIDDEN + h], acc);
}

extern "C" void kernel_launch(void* const* d_in, const int* in_sizes, int n_in,
                              void* d_out, int out_size, void* d_ws, size_t ws_size,
                              hipStream_t stream) {
    const float* x     = (const float*)d_in[0];
    const int*   ei    = (const int*)d_in[1];   // [2, E] row-major: src then dst
    const int*   batch = (const int*)d_in[2];
    const float* Wrel  = (const float*)d_in[3];
    const float* brel  = (const float*)d_in[4];
    const float* Wroot = (const float*)d_in[5];
    float*       gx    = (float*)d_out;

    const int n = in_sizes[0] / HIDDEN;
    const int e = in_sizes[1] / 2;
    const int g = out_size / HIDDEN;

    // workspace layout: p_rel[n] | p_root[n] | sval[n] | gmax[g] | denom[g]
    float*    prel  = (float*)d_ws;
    float*    proot = prel + n;
    float*    sval  = proot + n;
    unsigned* gmax  = (unsigned*)(sval + n);
    float*    denom = (float*)(gmax + g);

    const int* src = ei;
    const int* dst = ei + e;

    int total = n > g * HIDDEN ? n : g * HIDDEN;
    gap_init<<<(total + 255) / 256, 256, 0, stream>>>(sval, gmax, denom, gx, n, g);

    gap_proj<<<(n + 127) / 128, 256, 0, stream>>>(x, Wrel, Wroot, prel, proot, n);

    gap_edges<<<(e + 255) / 256, 256, 0, stream>>>(src, dst, prel, sval, e);

    gap_score<<<(n + 255) / 256, 256, 0, stream>>>(sval, proot, brel, batch, gmax, n);

    gap_exp<<<(n + 255) / 256, 256, 0, stream>>>(sval, batch, gmax, denom, n);

    gap_pool<<<(n + 31) / 32, HIDDEN, 0, stream>>>(x, batch, sval, denom, gx, n);
}